// BPLoss_16698832847232
// MI455X (gfx1250) — compile-verified
//
#include <hip/hip_runtime.h>

#define NROW 4096
#define BITD 64
#define LBL  10

typedef float v2f __attribute__((ext_vector_type(2)));
typedef float v8f __attribute__((ext_vector_type(8)));

// ---------- order-preserving float <-> u32 key mapping ----------
__device__ __forceinline__ unsigned ordf(float f) {
  unsigned u = __float_as_uint(f);
  return u ^ ((u & 0x80000000u) ? 0xFFFFFFFFu : 0x80000000u);
}
__device__ __forceinline__ float invord(unsigned k) {
  unsigned u = (k & 0x80000000u) ? (k ^ 0x80000000u) : ~k;
  return __uint_as_float(u);
}

// ---------- deterministic wave32 reductions ----------
__device__ __forceinline__ float wredf(float v) {
#pragma unroll
  for (int o = 16; o > 0; o >>= 1) v += __shfl_xor(v, o, 32);
  return v;
}
__device__ __forceinline__ int wredi(int v) {
#pragma unroll
  for (int o = 16; o > 0; o >>= 1) v += __shfl_xor(v, o, 32);
  return v;
}

// MODE 0: find k-th smallest key among similar (ascending values)
// MODE 1: find k-th smallest key among dissimilar with key = ~ord(x) (descending values)
// All waves call this with identical trip counts -> __syncthreads is uniform.
template <int MODE>
__device__ unsigned radix_kth(const float* __restrict__ row,
                              const unsigned* __restrict__ yb, unsigned ybi,
                              int k, unsigned* __restrict__ h,
                              unsigned* __restrict__ bc, int lane) {
  unsigned prefix = 0;
  int remaining = k;
#pragma unroll
  for (int level = 0; level < 4; ++level) {
    const int shift = 24 - 8 * level;
    const unsigned hmask = (level == 0) ? 0u : (0xFFFFFFFFu << (shift + 8));
    for (int t = lane; t < 256; t += 32) h[t] = 0u;
    __syncthreads();
    for (int j = lane; j < NROW; j += 32) {
      float x = row[j];
      bool sim = (ybi & yb[j]) != 0u;
      unsigned key;
      if (MODE == 0) key = sim ? ordf(x) : 0xFFFFFFFFu;
      else           key = sim ? 0xFFFFFFFFu : ~ordf(x);
      if ((key & hmask) == prefix)
        atomicAdd(&h[(key >> shift) & 255u], 1u);
    }
    __syncthreads();
    if (lane == 0) {
      int rem = remaining;
      unsigned b = 0;
      for (; b < 256u; ++b) {
        int cb = (int)h[b];
        if (cb >= rem) break;
        rem -= cb;
      }
      if (b > 255u) b = 255u;
      bc[0] = prefix | (b << shift);
      bc[1] = (unsigned)rem;
    }
    __syncthreads();
    prefix = bc[0];
    remaining = (int)bc[1];
  }
  return prefix;  // exact 32-bit key of the k-th element
}

// ---------- kernel 0: pack label rows into bitmasks ----------
__global__ void ybits_kernel(const int* __restrict__ y, unsigned* __restrict__ yb) {
  int i = blockIdx.x * blockDim.x + threadIdx.x;
  if (i < NROW) {
    unsigned m = 0u;
#pragma unroll
    for (int l = 0; l < LBL; ++l) m |= (y[i * LBL + l] != 0) ? (1u << l) : 0u;
    yb[i] = m;
  }
}

// ---------- fused kernel: WMMA GEMM slab in LDS + radix-select + loss ----------
__global__ __launch_bounds__(256) void bploss_fused(
    const float* __restrict__ u, const float* __restrict__ v,
    const unsigned* __restrict__ ybits, float* __restrict__ rowLoss,
    float* __restrict__ rowValid) {
  extern __shared__ char smemraw[];
  float* rowbuf = (float*)smemraw;                    // 16*4096 f32 = 256 KB
  float* uLDS = rowbuf + 16 * NROW;                   // 16*64 f32  = 4 KB
  unsigned* ybLDS = (unsigned*)(uLDS + 16 * BITD);    // 4096 u32   = 16 KB
  unsigned* hist = ybLDS + NROW;                      // 8*256 u32  = 8 KB
  unsigned* bcast = hist + 8 * 256;                   // 8*2 u32

  const int tid = threadIdx.x;
  const int wave = tid >> 5;
  const int lane = tid & 31;
  const int i0 = blockIdx.x * 16;

  for (int t = tid; t < 16 * BITD; t += 256) uLDS[t] = u[i0 * BITD + t];
  for (int t = tid; t < NROW; t += 256) ybLDS[t] = ybits[t];
  __syncthreads();

  // ---- Phase A: rowbuf[r][j] = dot(u[i0+r], v[j]) via V_WMMA_F32_16X16X4_F32
  const int m = lane & 15;
  const int h2 = (lane >> 4) * 2;  // K sub-offset per ISA 16x4 f32 A layout
  const float* urow = uLDS + m * BITD;
  for (int jt = wave; jt < NROW / 16; jt += 8) {
    const int j0 = jt * 16;
    const float* vrow = v + (size_t)(j0 + m) * BITD;
    v8f acc = {0.f, 0.f, 0.f, 0.f, 0.f, 0.f, 0.f, 0.f};
#pragma unroll
    for (int kk = 0; kk < BITD; kk += 4) {
      v2f a, b;
      a.x = urow[kk + h2];
      a.y = urow[kk + h2 + 1];
      b.x = vrow[kk + h2];
      b.y = vrow[kk + h2 + 1];
      acc = __builtin_amdgcn_wmma_f32_16x16x4_f32(false, a, false, b, (short)0,
                                                  acc, false, false);
    }
    const int mb = (lane >> 4) * 8;  // C/D layout: lanes16-31 hold M=8..15
#pragma unroll
    for (int r = 0; r < 8; ++r) rowbuf[(mb + r) * NROW + j0 + m] = acc[r];
  }
  __syncthreads();

  // ---- Phase B: per-row statistics (wave handles rows wave and wave+8)
  unsigned* h = hist + wave * 256;
  unsigned* bc = bcast + wave * 2;
  const float upper = (float)BITD / 4.0f;           // 16
  const float cc = -(3.0f / 32.0f) * logf(99.0f);   // (1/right)*ln(1/99)
  const float ac = 2.0f * cc;                       // a == 2 analytically

  for (int rr = 0; rr < 2; ++rr) {
    const int rloc = wave + rr * 8;
    const int i = i0 + rloc;
    const unsigned ybi = ybLDS[i];
    const float* row = rowbuf + rloc * NROW;

    // pass 1: counts + means
    int ns_l = 0;
    float sS = 0.f, sD = 0.f;
    for (int j = lane; j < NROW; j += 32) {
      float x = row[j];
      if ((ybi & ybLDS[j]) != 0u) { ns_l++; sS += x; }
      else sD += x;
    }
    int ns = wredi(ns_l);
    sS = wredf(sS);
    sD = wredf(sD);
    int nd = NROW - ns;
    float meanS = fminf(fmaxf(sS / fmaxf((float)ns, 1.f), 0.f), upper);
    float meanDS = fminf(fmaxf(sD / fmaxf((float)nd, 1.f), 0.f), upper);

    int kS = ns - (ns * 9) / 10; if (kS < 1) kS = 1;  // tail size, >=1
    int kD = nd - (nd * 9) / 10; if (kD < 1) kD = 1;

    unsigned Ks = radix_kth<0>(row, ybLDS, ybi, kS, h, bc, lane);
    unsigned Kd = radix_kth<1>(row, ybLDS, ybi, kD, h, bc, lane);

    // deterministic tail sums with exact tie handling
    float s0 = 0.f, s1 = 0.f;
    int c0 = 0, c1 = 0;
    for (int j = lane; j < NROW; j += 32) {
      float x = row[j];
      bool sim = (ybi & ybLDS[j]) != 0u;
      unsigned k0 = sim ? ordf(x) : 0xFFFFFFFFu;
      unsigned k1 = sim ? 0xFFFFFFFFu : ~ordf(x);
      if (k0 < Ks) { s0 += x; c0++; }
      if (k1 < Kd) { s1 += x; c1++; }
    }
    s0 = wredf(s0); c0 = wredi(c0);
    s1 = wredf(s1); c1 = wredi(c1);
    float similarMin = (s0 + (float)(kS - c0) * invord(Ks)) / (float)kS;
    float dissimilarMax = (s1 + (float)(kD - c1) * invord(~Kd)) / (float)kD;

    float BP = meanS - (upper - meanS) / upper * fabsf(meanS - dissimilarMax);
    float BPds = meanDS - meanDS / upper * fabsf(meanDS - similarMin);
    float d = -cc * BP, g = -ac * BP;        // base = ln(1) = 0
    float d2 = -cc * BPds, g2 = -ac * BPds;

    // pass: piecewise-linear + softplus
    float pos = 0.f, nav = 0.f;
    for (int j = lane; j < NROW; j += 32) {
      float x = row[j];
      bool sim = (ybi & ybLDS[j]) != 0u;
      if (sim) {
        float f = (x > BP) ? (cc * x + d) : (ac * x + g);
        pos += fmaxf(f, 0.f) + log1pf(expf(-fabsf(f)));
      } else {
        float f = (x < BPds) ? (cc * x + d2) : (ac * x + g2);
        float nf = -f;
        nav += fmaxf(nf, 0.f) + log1pf(expf(-fabsf(nf)));
      }
    }
    pos = wredf(pos);
    nav = wredf(nav);
    float vf = (ns > 0 && nd > 0) ? 1.f : 0.f;
    if (lane == 0) {
      float loss = pos / fmaxf((float)ns, 1.f) + nav / fmaxf((float)nd, 1.f);
      rowLoss[i] = (vf != 0.f) ? loss : 0.f;  // keep NaN of invalid rows out
      rowValid[i] = vf;
    }
  }
}

// ---------- kernel 2: deterministic final reduction ----------
__global__ __launch_bounds__(256) void finalize_kernel(
    const float* __restrict__ rowLoss, const float* __restrict__ rowValid,
    float* __restrict__ out) {
  __shared__ float sL[256], sV[256];
  int t = threadIdx.x;
  float l = 0.f, c = 0.f;
  for (int j = t; j < NROW; j += 256) { l += rowLoss[j]; c += rowValid[j]; }
  sL[t] = l;
  sV[t] = c;
  __syncthreads();
  for (int o = 128; o > 0; o >>= 1) {
    if (t < o) { sL[t] += sL[t + o]; sV[t] += sV[t + o]; }
    __syncthreads();
  }
  if (t == 0) out[0] = (sV[0] > 0.f) ? (sL[0] / fmaxf(sV[0], 1.f)) : 0.f;
}

extern "C" void kernel_launch(void* const* d_in, const int* in_sizes, int n_in,
                              void* d_out, int out_size, void* d_ws,
                              size_t ws_size, hipStream_t stream) {
  (void)in_sizes; (void)n_in; (void)out_size; (void)ws_size;
  const float* u = (const float*)d_in[0];
  const float* v = (const float*)d_in[1];
  const int* y = (const int*)d_in[2];
  float* out = (float*)d_out;

  float* rowLoss = (float*)d_ws;              // 4096 f32
  float* rowValid = rowLoss + NROW;           // 4096 f32
  unsigned* yb = (unsigned*)(rowValid + NROW);// 4096 u32  (48 KB total)

  ybits_kernel<<<NROW / 256, 256, 0, stream>>>(y, yb);

  size_t lds = (size_t)(16 * NROW + 16 * BITD) * 4 +
               (size_t)(NROW + 8 * 256 + 16) * 4;  // ~284 KB < 320 KB WGP LDS
  hipFuncSetAttribute(reinterpret_cast<const void*>(bploss_fused),
                      hipFuncAttributeMaxDynamicSharedMemorySize, (int)lds);
  bploss_fused<<<NROW / 16, 256, lds, stream>>>(u, v, yb, rowLoss, rowValid);

  finalize_kernel<<<1, 256, 0, stream>>>(rowLoss, rowValid, out);
}